// STModel_43035572306761
// MI455X (gfx1250) — compile-verified
//
#include <hip/hip_runtime.h>
#include <hip/hip_bf16.h>

// ---------------------------------------------------------------------------
// ST-GCN fused pipeline for MI455X (gfx1250, wave32).
//
//   K1  deg init   : deg[n] = 1 (self loop)
//   K2  deg edges  : atomicAdd(deg[dst], w)
//   K3  finalize   : dinv = rsqrt(deg); transpose x [48,N]->[N,48];
//                    aggt[n,:] = x_t[n,:] * dinv[n]^2   (self-loop folded in)
//   K4  aggregate  : per wave = 1 edge; lane = time slice (48 contiguous)
//                    aggt[dst,s] += x_t[src,s] * dinv[src]*w*dinv[dst]
//   K5  epilogue   : per wave = 1 (b,n): two V_WMMA_F32_16X16X4_F32 doing
//                    z[t,co] = A(t,k) * w2(k,co) + Cbias(t,co), then
//                    ReLU -> mean_t -> dot(lin_w) via shfl_xor reductions.
//                    All lane predicates hoisted out of the loop: loads are
//                    unconditional (clamped offset * 0/1 mask) so EXEC stays
//                    full and no per-iteration exec save/restore is emitted.
// ---------------------------------------------------------------------------

typedef __attribute__((ext_vector_type(2))) float v2f;
typedef __attribute__((ext_vector_type(8))) float v8f;

#define BT 48
#define TT 12
#define BB 4
#define HCH 32

__global__ void k_deg_init(float* __restrict__ deg, int n_nodes) {
    int n = blockIdx.x * blockDim.x + threadIdx.x;
    if (n < n_nodes) deg[n] = 1.0f;   // self-loop weight 1.0
}

__global__ void k_deg_edges(const int* __restrict__ ei, const float* __restrict__ ew,
                            float* __restrict__ deg, int n_edges) {
    int e = blockIdx.x * blockDim.x + threadIdx.x;
    if (e < n_edges) {
        int d = ei[(size_t)n_edges + e];
        atomicAdd(&deg[d], ew[e]);
    }
}

// dinv = deg^-1/2 ; transpose x[48,N] -> xt[N,48] ; aggt init = x * dinv^2 (self loop)
__global__ void k_finalize(const float* __restrict__ x, float* __restrict__ dinv,
                           float* __restrict__ xt, float* __restrict__ aggt, int n_nodes) {
    int n = blockIdx.x * blockDim.x + threadIdx.x;
    if (n >= n_nodes) return;
    float deg = dinv[n];
    float di  = deg > 0.0f ? rsqrtf(deg) : 0.0f;
    dinv[n] = di;
    float sn = di * di;                 // self-loop norm = dinv*1*dinv
    float buf[BT];
#pragma unroll
    for (int s = 0; s < BT; ++s)        // coalesced column reads
        buf[s] = x[(size_t)s * n_nodes + n];
    float4* xt4 = (float4*)(xt + (size_t)n * BT);    // n*192B: 16B aligned
    float4* ag4 = (float4*)(aggt + (size_t)n * BT);
#pragma unroll
    for (int q = 0; q < BT / 4; ++q) {
        float4 v = make_float4(buf[4*q], buf[4*q+1], buf[4*q+2], buf[4*q+3]);
        xt4[q] = v;
        ag4[q] = make_float4(v.x * sn, v.y * sn, v.z * sn, v.w * sn);
    }
}

// One wave per edge; lanes cover the 48 time slices (contiguous -> coalesced
// gathers and per-line-batched L2 atomics). Edge scalars are wave-uniform.
__global__ void k_aggregate(const int* __restrict__ ei, const float* __restrict__ ew,
                            const float* __restrict__ dinv, const float* __restrict__ xt,
                            float* __restrict__ aggt, int n_edges) {
    int lane = threadIdx.x & 31;
    long long wid = (long long)blockIdx.x * (blockDim.x >> 5) + (threadIdx.x >> 5);
    long long nw  = (long long)gridDim.x * (blockDim.x >> 5);
    for (long long e = wid; e < n_edges; e += nw) {
        int   s_  = ei[e];
        int   d_  = ei[(size_t)n_edges + e];
        float w   = ew[e];
        float nrm = dinv[s_] * w * dinv[d_];
        const float* xs = xt   + (size_t)s_ * BT;
        float*       ad = aggt + (size_t)d_ * BT;
        atomicAdd(&ad[lane], xs[lane] * nrm);
        if (lane < 16)
            atomicAdd(&ad[32 + lane], xs[32 + lane] * nrm);
    }
}

// WMMA epilogue: one (b,n) per wave per iteration.
//   A(16x4)[t,k] = agg_pad[t+k-1]   (t rows 12..15 zero; B row k=3 zero)
//   B(4x16)[k,co]= w2[k,co] = sum_ci gcn_w[ci]*conv_w[co,ci,k]
//   C(16x16)[t,co] = conv_b[co] + sum over in-bounds taps of bb[k,co],
//                    bb[k,co] = sum_ci gcn_b[ci]*conv_w[co,ci,k]
// D = A*B + C is the pre-ReLU conv output. Rows t>=12 of D are exactly zero
// (A rows zero, C rows zero), so ReLU-accumulation sums all 8 held rows
// unconditionally. Then mean_t(relu(D)) . lin_w + lin_b.
__global__ void k_epilogue(const float* __restrict__ aggt,
                           const float* __restrict__ gw, const float* __restrict__ gb,
                           const float* __restrict__ cw, const float* __restrict__ cb,
                           const float* __restrict__ lwp, const float* __restrict__ lbp,
                           float* __restrict__ out, int n_nodes) {
    const int lane = threadIdx.x & 31;
    const int h    = lane >> 4;      // lane half
    const int m    = lane & 15;      // row (A) / column (B,C,D) within half

    // ---- per-wave constant setup (amortized over the node loop) ----
    // A-layout: lane holds A[m, 2h+v] in vgpr v ; B-layout: lane holds B[2h+v, m].
    v2f bmat0, bmat1;                // co = m  and co = m+16
#pragma unroll
    for (int v = 0; v < 2; ++v) {
        int k = 2 * h + v;
        float a0 = 0.0f, a1 = 0.0f;
        if (k < 3) {
            for (int ci = 0; ci < HCH; ++ci) {
                float g = gw[ci];
                a0 += g * cw[(size_t)m        * 96 + ci * 3 + k];
                a1 += g * cw[(size_t)(m + 16) * 96 + ci * 3 + k];
            }
        }
        bmat0[v] = a0;
        bmat1[v] = a1;
    }
    float bbA[3], bbB[3];
#pragma unroll
    for (int k = 0; k < 3; ++k) {
        float a0 = 0.0f, a1 = 0.0f;
        for (int ci = 0; ci < HCH; ++ci) {
            float g = gb[ci];
            a0 += g * cw[(size_t)m        * 96 + ci * 3 + k];
            a1 += g * cw[(size_t)(m + 16) * 96 + ci * 3 + k];
        }
        bbA[k] = a0;
        bbB[k] = a1;
    }
    v8f cmat0, cmat1;                // C-layout: vgpr j holds row t = j + 8h, col m
#pragma unroll
    for (int j = 0; j < 8; ++j) {
        int t = j + 8 * h;
        float c0 = 0.0f, c1 = 0.0f;
        if (t < TT) {
            c0 = cb[m]      + bbA[1] + (t > 0 ? bbA[0] : 0.0f) + (t < TT - 1 ? bbA[2] : 0.0f);
            c1 = cb[m + 16] + bbB[1] + (t > 0 ? bbB[0] : 0.0f) + (t < TT - 1 ? bbB[2] : 0.0f);
        }
        cmat0[j] = c0;
        cmat1[j] = c1;
    }
    const float lwA = lwp[m], lwB = lwp[16 + m];
    const float lb  = lbp[0];

    // Loop-invariant per-lane gather offsets + 0/1 masks (no branches in loop).
    const int  i0  = m + 2 * h - 1;           // A[m, 2h]   = agg_pad[m + 2h - 1]
    const int  i1  = i0 + 1;                  // A[m, 2h+1] = agg_pad[m + 2h]
    const bool ok0 = (m < TT) && (i0 >= 0) && (i0 < TT);
    const bool ok1 = (m < TT) && (i1 < TT);
    const int  o0  = ok0 ? i0 : 0;
    const int  o1  = ok1 ? i1 : 0;
    const float msk0 = ok0 ? 1.0f : 0.0f;
    const float msk1 = ok1 ? 1.0f : 0.0f;

    const int wid = blockIdx.x * (blockDim.x >> 5) + (threadIdx.x >> 5);
    const int nw  = gridDim.x * (blockDim.x >> 5);

#pragma unroll
    for (int b = 0; b < BB; ++b) {
        const float* agb  = aggt + b * TT;
        float*       outb = out + (size_t)b * n_nodes;
        for (int n = wid; n < n_nodes; n += nw) {   // uniform loop: EXEC stays full
            const float* ag = agb + (size_t)n * BT; // 12 contiguous slices

            v2f amat;
            amat[0] = ag[o0] * msk0;                // unconditional clamped gathers
            amat[1] = ag[o1] * msk1;

            v8f d0 = __builtin_amdgcn_wmma_f32_16x16x4_f32(false, amat, false, bmat0,
                                                           (short)0, cmat0, false, false);
            v8f d1 = __builtin_amdgcn_wmma_f32_16x16x4_f32(false, amat, false, bmat1,
                                                           (short)0, cmat1, false, false);

            // ReLU + sum over all 8 held rows (rows t>=12 are exactly 0).
            float sA = 0.0f, sB = 0.0f;
#pragma unroll
            for (int j = 0; j < 8; ++j) {
                sA += fmaxf(d0[j], 0.0f);
                sB += fmaxf(d1[j], 0.0f);
            }
            sA += __shfl_xor(sA, 16);    // combine row halves (same co)
            sB += __shfl_xor(sB, 16);

            float v = sA * lwA + sB * lwB;           // weight by lin_w
            v += __shfl_xor(v, 1);
            v += __shfl_xor(v, 2);
            v += __shfl_xor(v, 4);
            v += __shfl_xor(v, 8);                   // sum over the 16 co-lanes

            if (lane == 0)
                outb[n] = v * (1.0f / (float)TT) + lb;  // mean over T, + lin_b
        }
    }
}

extern "C" void kernel_launch(void* const* d_in, const int* in_sizes, int n_in,
                              void* d_out, int out_size, void* d_ws, size_t ws_size,
                              hipStream_t stream) {
    const float* x  = (const float*)d_in[0];   // [B,T,N] f32
    const int*   ei = (const int*)d_in[1];     // [2,E]   int
    const float* ew = (const float*)d_in[2];   // [E]
    const float* gw = (const float*)d_in[3];   // [1,32]
    const float* gb = (const float*)d_in[4];   // [32]
    const float* cw = (const float*)d_in[5];   // [32,32,3]
    const float* cb = (const float*)d_in[6];   // [32]
    const float* lw = (const float*)d_in[7];   // [32,1]
    const float* lb = (const float*)d_in[8];   // [1]
    float* out = (float*)d_out;

    const int E = in_sizes[2];
    const int N = in_sizes[0] / BT;

    float* dinv = (float*)d_ws;                 // N floats (deg, then dinv in place)
    float* xt   = dinv + N;                     // N*48 floats
    float* aggt = xt + (size_t)N * BT;          // N*48 floats

    k_deg_init <<<(N + 255) / 256, 256, 0, stream>>>(dinv, N);
    k_deg_edges<<<(E + 255) / 256, 256, 0, stream>>>(ei, ew, dinv, E);
    k_finalize <<<(N + 255) / 256, 256, 0, stream>>>(x, dinv, xt, aggt, N);

    int aggBlocks = (E + 7) / 8;                // 8 waves/block, 1 edge/wave
    k_aggregate<<<aggBlocks, 256, 0, stream>>>(ei, ew, dinv, xt, aggt, E);

    k_epilogue <<<1024, 256, 0, stream>>>(aggt, gw, gb, cw, cb, lw, lb, out, N);
}